// CnnPatchDropout_21784074126113
// MI455X (gfx1250) — compile-verified
//
#include <hip/hip_runtime.h>
#include <cstdint>
#include <cstddef>

// ---------------------------------------------------------------------------
// Problem constants (mirror the reference)
// ---------------------------------------------------------------------------
#define N_B   64
#define N_C   64
#define N_H   128
#define N_W   128
#define N_HP  16
#define N_WP  16
#define N_PQ  256   // patches per image
// P_DROP = 0.5, frac in [0.1, 0.5]

typedef float v4f __attribute__((ext_vector_type(4)));
typedef int   v4i __attribute__((ext_vector_type(4)));

// ---------------------------------------------------------------------------
// Threefry-2x32 (JAX's PRNG), usable constexpr on host and inline on device.
// ---------------------------------------------------------------------------
struct TFOut { uint32_t a, b; };

__host__ __device__ constexpr uint32_t rotl32(uint32_t v, int n) {
  return (v << n) | (v >> (32 - n));
}

__host__ __device__ constexpr TFOut threefry2x32(uint32_t k0, uint32_t k1,
                                                 uint32_t x0, uint32_t x1) {
  uint32_t ks[3] = { k0, k1, k0 ^ k1 ^ 0x1BD11BDAu };
  const int R[2][4] = { {13, 15, 26, 6}, {17, 29, 16, 24} };
  x0 += ks[0]; x1 += ks[1];
  for (int i = 0; i < 5; ++i) {
    for (int j = 0; j < 4; ++j) {
      x0 += x1;
      x1 = rotl32(x1, R[i & 1][j]);
      x1 ^= x0;
    }
    x0 += ks[(i + 1) % 3];
    x1 += ks[(i + 2) % 3] + (uint32_t)(i + 1);
  }
  return { x0, x1 };
}

// JAX classic random_bits for a flat array of n 32-bit draws:
// counts = iota(n); x0 = counts[0:n/2], x1 = counts[n/2:n]; bits = concat(o0, o1).
__device__ inline uint32_t jax_bits(uint32_t k0, uint32_t k1,
                                    uint32_t idx, uint32_t n) {
  const uint32_t half = n >> 1;
  if (idx < half) {
    return threefry2x32(k0, k1, idx, idx + half).a;
  } else {
    return threefry2x32(k0, k1, idx - half, idx).b;
  }
}

__device__ inline float jax_u01(uint32_t bits) {
  // (bits >> 9) | 0x3f800000 viewed as float, minus 1.0  ->  [0, 1)
  return __uint_as_float(0x3f800000u | (bits >> 9)) - 1.0f;
}

// ---------------------------------------------------------------------------
// Kernel A (emitted first so the disasm snippet shows it): streaming apply.
// One block = one contiguous 4KB strip (b, c, hp): 8 rows x 128 floats.
// 256 threads x float4. Embed slice (16 patches x 256B) is staged to LDS
// with gfx1250 async-to-LDS loads only when the block has a masked patch.
// ---------------------------------------------------------------------------
#if defined(__AMDGCN__) && __has_builtin(__builtin_amdgcn_global_load_async_to_lds_b128)
#define USE_ASYNC_LDS 1
#else
#define USE_ASYNC_LDS 0
#endif

__global__ __launch_bounds__(256) void pd_apply(
    const float* __restrict__ batch,
    const float* __restrict__ embed,
    const uint8_t* __restrict__ mask,
    float* __restrict__ out) {
  __shared__ float se[N_WP * 64];  // 16 patches x 64 floats = 4 KB

  const uint32_t blk = blockIdx.x;
  const uint32_t hp  = blk & (N_HP - 1);
  const uint32_t c   = (blk >> 4) & (N_C - 1);
  const uint32_t b   = blk >> 10;
  const uint32_t t   = threadIdx.x;

  // contiguous strip: ((b*C + c)*H + hp*8) * W
  const size_t strip = (size_t)(b * N_C + c) * (N_H * N_W) + (size_t)hp * (8 * N_W);

  const uint32_t w   = (t * 4u) & (N_W - 1);  // column within the 128-wide row
  const uint32_t kh  = t >> 5;                // row within the patch (0..7)
  const uint32_t wp  = w >> 3;                // which patch along W (0..15)

  const bool masked = mask[(size_t)b * N_PQ + hp * N_WP + wp] != 0;
  const bool any    = __syncthreads_or((int)masked) != 0;  // uniform across block

  if (any) {
    // stage embed[(hp*16+lwp), c, :, :] (256 B per patch) into LDS, coalesced:
    // thread t -> patch lwp = t>>4, 16-byte chunk sub = t&15
    const uint32_t lwp = t >> 4;
    const uint32_t sub = t & 15u;
    const float* g = embed + ((size_t)(hp * N_WP + lwp) * N_C + c) * 64 + sub * 4u;
    float* l = &se[lwp * 64u + sub * 4u];
#if USE_ASYNC_LDS
    __builtin_amdgcn_global_load_async_to_lds_b128(
        (__attribute__((address_space(1))) v4i*)g,
        (__attribute__((address_space(3))) v4i*)l,
        0, 0);
#if __has_builtin(__builtin_amdgcn_s_wait_asynccnt)
    __builtin_amdgcn_s_wait_asynccnt(0);
#else
    asm volatile("s_wait_asynccnt 0" ::: "memory");
#endif
#else
    *reinterpret_cast<v4f*>(l) = *reinterpret_cast<const v4f*>(g);
#endif
    __syncthreads();
  }

  v4f v;
  if (masked) {
    // patch row from LDS (16B aligned: w&7 is 0 or 4)
    v = *reinterpret_cast<const v4f*>(&se[wp * 64u + kh * 8u + (w & 7u)]);
  } else {
    // read-once stream: non-temporal so it doesn't evict the resident embed
    v = __builtin_nontemporal_load(
        reinterpret_cast<const v4f*>(batch + strip + (size_t)t * 4u));
  }
  __builtin_nontemporal_store(
      v, reinterpret_cast<v4f*>(out + strip + (size_t)t * 4u));
}

// ---------------------------------------------------------------------------
// Kernel B: compute the [B, PQ] drop mask into d_ws (one byte per patch).
// rank(i) = #{j : r[j] < r[i]  or (r[j]==r[i] and j<i)}  == argsort(argsort(r))
// ---------------------------------------------------------------------------
__global__ __launch_bounds__(N_PQ) void mask_kernel(
    uint8_t* __restrict__ mask,
    uint32_t k1a, uint32_t k1b,   // apply-coin key
    uint32_t k2a, uint32_t k2b,   // fraction key
    uint32_t k3a, uint32_t k3b) { // permutation key
  __shared__ float sr[N_PQ];
  const uint32_t b = blockIdx.x;
  const uint32_t i = threadIdx.x;
  const uint32_t idx = b * N_PQ + i;

  const float r = jax_u01(jax_bits(k3a, k3b, idx, N_B * N_PQ));
  sr[i] = r;

  // per-image scalars (uniform; compiler scalarizes these to SALU)
  const float u1 = jax_u01(jax_bits(k1a, k1b, b, N_B));
  const float u2 = jax_u01(jax_bits(k2a, k2b, b, N_B));
  const bool  apply = u1 < 0.5f;
  const int   dropped = (int)floorf((u2 * 0.4f + 0.1f) * (float)N_PQ);

  __syncthreads();
  int cnt = 0;
#pragma unroll 8
  for (int j = 0; j < N_PQ; ++j) {
    const float v = sr[j];
    cnt += (v < r) || (v == r && j < (int)i);
  }
  mask[idx] = (uint8_t)((cnt < dropped) && apply);
}

// ---------------------------------------------------------------------------
// Launch
// ---------------------------------------------------------------------------
extern "C" void kernel_launch(void* const* d_in, const int* in_sizes, int n_in,
                              void* d_out, int out_size, void* d_ws, size_t ws_size,
                              hipStream_t stream) {
  const float* batch = (const float*)d_in[0];
  const float* embed = (const float*)d_in[1];
  float* out = (float*)d_out;
  uint8_t* mask = (uint8_t*)d_ws;  // needs B*PQ = 16384 bytes

  // jax.random.split(key(42), 3) folded at compile time:
  // counts = iota(6): lanes (0,3), (1,4), (2,5); bits = concat(o0, o1);
  // k1 = (f0.a, f1.a), k2 = (f2.a, f0.b), k3 = (f1.b, f2.b)
  constexpr TFOut f0 = threefry2x32(0u, 42u, 0u, 3u);
  constexpr TFOut f1 = threefry2x32(0u, 42u, 1u, 4u);
  constexpr TFOut f2 = threefry2x32(0u, 42u, 2u, 5u);

  hipLaunchKernelGGL(mask_kernel, dim3(N_B), dim3(N_PQ), 0, stream,
                     mask, f0.a, f1.a, f2.a, f0.b, f1.b, f2.b);

  hipLaunchKernelGGL(pd_apply, dim3(N_B * N_C * N_HP), dim3(256), 0, stream,
                     batch, embed, mask, out);
}